// SSDTableBatchedEmbeddingBags_13305808683301
// MI455X (gfx1250) — compile-verified
//
#include <hip/hip_runtime.h>

// Table-batched embedding-bag forward (TBE), CDNA5 / gfx1250.
// T=8 tables, ROWS=100000, D=128, B=8192 bags/table, L=32 avg bag len.
// out[b, t*D + d] = sum_{j in bag(t,b)} weights[indices[j] + t*ROWS, d]

#define T_TABLES 8
#define N_ROWS   100000
#define EMB_D    128
#define N_BAGS   8192
#define WAVES_PER_BLOCK 8
#define ROW_BYTES (EMB_D * 4)          // 512 B per row

typedef __attribute__((ext_vector_type(4))) float v4f;

__global__ __launch_bounds__(WAVES_PER_BLOCK * 32)
void tbe_forward_kernel(const int* __restrict__ indices,
                        const int* __restrict__ offsets,
                        const float* __restrict__ weights,
                        float* __restrict__ out)
{
    const int lane = threadIdx.x & 31;
    const int wave = threadIdx.x >> 5;
    const int bag  = blockIdx.x * WAVES_PER_BLOCK + wave;   // segment id in [0, T*B)
    if (bag >= T_TABLES * N_BAGS) return;

    const int t = bag >> 13;             // bag / 8192  (table id)
    const int b = bag & (N_BAGS - 1);    // bag % 8192

    const int start = offsets[bag];
    const int end   = offsets[bag + 1];

    // All byte offsets into the 409.6 MB weight array fit in unsigned 32-bit.
    const unsigned table_off = (unsigned)(t * N_ROWS) * (unsigned)ROW_BYTES;
    const char* __restrict__ wbytes = (const char*)weights;
    const unsigned lane_off = (unsigned)lane * 16u;   // this lane's float4 column

    v4f acc = (v4f)(0.0f);

    for (int jbase = start; jbase < end; jbase += 32) {
        const int rem = end - jbase;
        const int n   = rem < 32 ? rem : 32;

        // lane j holds the byte offset of the j-th row (one coalesced 128B load)
        unsigned my_off = 0;
        if (lane < n) {
            const int idx = indices[jbase + lane];
            my_off = table_off + ((unsigned)idx << 9);   // idx * 512
            // gfx1250 prefetch path: lane j prefetches all 4 cachelines of its
            // row; 4 global_prefetch_b8 instructions cover the whole 16KB bag.
            const char* rowp = wbytes + my_off;
            __builtin_prefetch(rowp,       0, 0);
            __builtin_prefetch(rowp + 128, 0, 0);
            __builtin_prefetch(rowp + 256, 0, 0);
            __builtin_prefetch(rowp + 384, 0, 0);
        }

        if (n == 32) {
            // Fast path (L == 32): v_readlane_b32 -> SGPR row offset, so each
            // gather is global_load_b128 with scalar base + v(lane*16).
            #pragma unroll
            for (int j = 0; j < 32; ++j) {
                const unsigned roff = __builtin_amdgcn_readlane(my_off, j);
                const v4f v = *(const v4f*)(wbytes + roff + lane_off);
                acc += v;
            }
        } else {
            for (int j = 0; j < n; ++j) {
                const unsigned roff = __builtin_amdgcn_readlane(my_off, j);
                const v4f v = *(const v4f*)(wbytes + roff + lane_off);
                acc += v;
            }
        }
    }

    // out is [B, T*D]; row stride = T*D = 1024 floats.
    // Streaming write-once output: non-temporal b128 store keeps L2 for
    // weight-row reuse (~2.6x average touches per row).
    char* obytes = (char*)out;
    const unsigned long long opos =
        (unsigned long long)b * (T_TABLES * ROW_BYTES) +
        (unsigned)t * ROW_BYTES + lane_off;
    __builtin_nontemporal_store(acc, (v4f*)(obytes + opos));
}

extern "C" void kernel_launch(void* const* d_in, const int* in_sizes, int n_in,
                              void* d_out, int out_size, void* d_ws, size_t ws_size,
                              hipStream_t stream)
{
    (void)in_sizes; (void)n_in; (void)out_size; (void)d_ws; (void)ws_size;

    const int*   indices = (const int*)d_in[0];     // [N]        int
    const int*   offsets = (const int*)d_in[1];     // [T*B+1]    int
    const float* weights = (const float*)d_in[2];   // [T*ROWS, D] fp32
    float*       out     = (float*)d_out;           // [B, T*D]   fp32

    const int total_bags = T_TABLES * N_BAGS;                                // 65536
    const int blocks = (total_bags + WAVES_PER_BLOCK - 1) / WAVES_PER_BLOCK; // 8192

    tbe_forward_kernel<<<blocks, WAVES_PER_BLOCK * 32, 0, stream>>>(
        indices, offsets, weights, out);
}